// CollectNeighboursAndEdgesToNodes_65249143160975
// MI455X (gfx1250) — compile-verified
//
#include <hip/hip_runtime.h>

// Problem constants from the reference (setup_inputs hard-codes these dims).
#define DN 256      // node feature dim
#define DE 128      // edge feature dim
#define KSLOTS 4    // max neighbour/edge slots per node
#define SENTINEL 0x7FFFFFFF

typedef float v4f __attribute__((ext_vector_type(4)));

// CDNA5 async global<->LDS copy path (ASYNCcnt-tracked), guarded so the file
// compiles on toolchains that lack the builtins (falls back to direct b128).
#if defined(__AMDGCN__) && defined(__gfx1250__)
#if __has_builtin(__builtin_amdgcn_global_load_async_to_lds_b128) && \
    __has_builtin(__builtin_amdgcn_global_store_async_from_lds_b128) && \
    __has_builtin(__builtin_amdgcn_s_wait_asynccnt)
#define USE_ASYNC_LDS 1
#endif
#endif
#ifndef USE_ASYNC_LDS
#define USE_ASYNC_LDS 0
#endif

#if USE_ASYNC_LDS
// Exact parameter types per the compiler diagnostic:
//   int __attribute__((vector_size(16))) * in AS1 (global) / AS3 (LDS).
typedef int b128_t __attribute__((vector_size(16)));
typedef b128_t __attribute__((address_space(1)))* gp_b128;
typedef b128_t __attribute__((address_space(3)))* lp_b128;
#define AS_G128(p) ((gp_b128)(p))
#define AS_L128(p) ((lp_b128)(p))
#endif

// ---------------------------------------------------------------------------
// Workspace layout (ints): [out_deg:N][in_deg:N][omin:4*N][imin:4*N] = 10*N
// omin[k*N + n] = (k)-th smallest outgoing edge id of node n (or SENTINEL)
// imin[k*N + n] = (k)-th smallest incoming edge id of node n (or SENTINEL)
// ---------------------------------------------------------------------------

__global__ void init_ws_kernel(int* __restrict__ ws, long long nzero, long long ntot) {
    long long i = (long long)blockIdx.x * blockDim.x + threadIdx.x;
    if (i < ntot) ws[i] = (i < nzero) ? 0 : SENTINEL;
}

// Round 0: degrees (atomicAdd, order-independent) + smallest edge id (atomicMin).
__global__ void deg_min0_kernel(const int* __restrict__ senders,
                                const int* __restrict__ receivers,
                                int* __restrict__ out_deg, int* __restrict__ in_deg,
                                int* __restrict__ omin0, int* __restrict__ imin0,
                                int E) {
    int e = blockIdx.x * blockDim.x + threadIdx.x;
    if (e >= E) return;
    int s = senders[e];
    int r = receivers[e];
    atomicAdd(&out_deg[s], 1);
    atomicAdd(&in_deg[r], 1);
    atomicMin(&omin0[s], e);
    atomicMin(&imin0[r], e);
}

// Round k (k=1..3): k-th smallest = min of edge ids strictly greater than the
// (k-1)-th smallest. atomicMin is commutative -> deterministic result.
__global__ void next_min_kernel(const int* __restrict__ senders,
                                const int* __restrict__ receivers,
                                const int* __restrict__ oprev, const int* __restrict__ iprev,
                                int* __restrict__ ocur, int* __restrict__ icur,
                                int E) {
    int e = blockIdx.x * blockDim.x + threadIdx.x;
    if (e >= E) return;
    int s = senders[e];
    int r = receivers[e];
    if (e > oprev[s]) atomicMin(&ocur[s], e);
    if (e > iprev[r]) atomicMin(&icur[r], e);
}

// One block per node, one wave32 per slot k (blockDim = 4*32 = 128).
// Slot semantics (matches reference exactly):
//   k <  out_deg           -> outgoing edge with rank k   (k<4 always holds)
//   k >= out_deg, r=k-deg  -> incoming edge with rank r, valid iff r < in_deg
__global__ __launch_bounds__(KSLOTS * 32)
void gather_kernel(const float* __restrict__ nodes,
                   const float* __restrict__ edges,
                   const int*   __restrict__ senders,
                   const int*   __restrict__ receivers,
                   const int*   __restrict__ out_deg,
                   const int*   __restrict__ in_deg,
                   const int*   __restrict__ omin,   // [4][N]
                   const int*   __restrict__ imin,   // [4][N]
                   float* __restrict__ out_nbr,      // [N, 4*DN]
                   float* __restrict__ out_edge,     // [N, 4*DE]
                   int N) {
    const int n    = blockIdx.x;
    const int k    = threadIdx.x >> 5;
    const int lane = threadIdx.x & 31;

    const int d = out_deg[n];
    bool valid = false;
    int eid = 0, nbr = 0;
    if (k < d) {                       // outgoing slot, rank k (omin[k] exists: d > k)
        eid = omin[(size_t)k * N + n];
        nbr = receivers[eid];
        valid = true;
    } else {                           // incoming slot, rank r = k - d (0..3)
        int r = k - d;
        if (r < in_deg[n]) {
            eid = imin[(size_t)r * N + n];
            nbr = senders[eid];
            valid = true;
        }
    }

    const v4f z = (v4f)0.0f;

    // ---- Node row: 256 floats = 64 v4f; 32 lanes -> 2 per lane. ----
    // Write-once streaming output (614 MB) -> non-temporal stores keep the
    // L2-resident nodes/edges tables from being evicted. Loads stay RT for reuse.
    v4f* __restrict__ ndst = (v4f*)(out_nbr + (size_t)n * (KSLOTS * DN) + (size_t)k * DN);
    if (valid) {
        const v4f* __restrict__ nsrc = (const v4f*)(nodes + (size_t)nbr * DN);
        v4f a = nsrc[lane];
        v4f b = nsrc[lane + 32];
        __builtin_nontemporal_store(a, ndst + lane);
        __builtin_nontemporal_store(b, ndst + lane + 32);
    } else {
        __builtin_nontemporal_store(z, ndst + lane);
        __builtin_nontemporal_store(z, ndst + lane + 32);
    }

    // ---- Edge row: 128 floats = 32 v4f; 1 per lane. ----
    float* __restrict__ edst = out_edge + (size_t)n * (KSLOTS * DE) + (size_t)k * DE;
#if USE_ASYNC_LDS
    // Route the copy through the CDNA5 async units: global -> LDS -> global,
    // 16B per lane, without staging through VGPRs. Both ops are ASYNCcnt-
    // tracked and wave-ordered; s_wait_asynccnt 0 makes the LDS data visible
    // before the async store reads it. S_ENDPGM's implicit wait drains the rest.
    __shared__ __align__(16) float ebuf[KSLOTS][DE];  // 4 waves x 512 B strips
    if (valid) {
        const float* esrc = edges + (size_t)eid * DE;
        __builtin_amdgcn_global_load_async_to_lds_b128(
            AS_G128((float*)esrc + lane * 4), AS_L128(&ebuf[k][lane * 4]), 0, 0);
        __builtin_amdgcn_s_wait_asynccnt(0);
        __builtin_amdgcn_global_store_async_from_lds_b128(
            AS_G128(edst + lane * 4), AS_L128(&ebuf[k][lane * 4]), 0, 0);
    } else {
        __builtin_nontemporal_store(z, (v4f*)edst + lane);
    }
#else
    if (valid) {
        const v4f* __restrict__ esrc = (const v4f*)(edges + (size_t)eid * DE);
        v4f c = esrc[lane];
        __builtin_nontemporal_store(c, (v4f*)edst + lane);
    } else {
        __builtin_nontemporal_store(z, (v4f*)edst + lane);
    }
#endif
}

extern "C" void kernel_launch(void* const* d_in, const int* in_sizes, int n_in,
                              void* d_out, int out_size, void* d_ws, size_t ws_size,
                              hipStream_t stream) {
    const float* nodes     = (const float*)d_in[0];
    const float* edges     = (const float*)d_in[1];
    const int*   senders   = (const int*)d_in[2];
    const int*   receivers = (const int*)d_in[3];

    const int N = in_sizes[0] / DN;   // 100000
    const int E = in_sizes[2];        // 200000
    if (N <= 0 || E <= 0) return;

    int* ws      = (int*)d_ws;
    int* out_deg = ws;
    int* in_deg  = ws + (size_t)N;
    int* omin    = ws + (size_t)2 * N;   // 4*N
    int* imin    = ws + (size_t)6 * N;   // 4*N

    // 1) Re-initialize workspace every call (ws is not re-poisoned by harness,
    //    and results must not depend on prior contents).
    {
        long long ntot  = 10LL * N;
        long long nzero = 2LL * N;
        int blocks = (int)((ntot + 255) / 256);
        init_ws_kernel<<<blocks, 256, 0, stream>>>(ws, nzero, ntot);
    }

    const int eblocks = (E + 255) / 256;

    // 2) Degrees + smallest edge id per node (both directions).
    deg_min0_kernel<<<eblocks, 256, 0, stream>>>(senders, receivers,
                                                 out_deg, in_deg,
                                                 omin, imin, E);

    // 3) 2nd/3rd/4th smallest via filtered atomicMin rounds (needs the previous
    //    round complete -> separate launches provide the global sync).
    for (int k = 1; k < KSLOTS; ++k) {
        next_min_kernel<<<eblocks, 256, 0, stream>>>(senders, receivers,
                                                     omin + (size_t)(k - 1) * N,
                                                     imin + (size_t)(k - 1) * N,
                                                     omin + (size_t)k * N,
                                                     imin + (size_t)k * N, E);
    }

    // 4) Resolve slots and copy rows. d_out = [nbrs: N*4*DN][edgs: N*4*DE].
    float* out_nbr  = (float*)d_out;
    float* out_edge = out_nbr + (size_t)N * (KSLOTS * DN);
    gather_kernel<<<N, KSLOTS * 32, 0, stream>>>(nodes, edges, senders, receivers,
                                                 out_deg, in_deg, omin, imin,
                                                 out_nbr, out_edge, N);
}